// SpatialTransformer_89498528514121
// MI455X (gfx1250) — compile-verified
//
#include <hip/hip_runtime.h>
#include <stdint.h>

// Spatial transformer: 2D bilinear warp of [B,C,D,H,W] fp32 by displacement
// field v[B,2,D,H,W].  Memory-bound gather op (~640 MB traffic, ~27us floor
// at 23.3 TB/s).  MI455X strategy:
//   - per 32x32 output tile, async-stage a 48x48 (8-pixel halo) x 4-channel
//     src tile into LDS via global_load_async_to_lds_b128 (gfx1250 async path,
//     ASYNCcnt), zero padding baked into the tile,
//   - sample the 4 bilinear corners from LDS (pairs fuse to ds_load_2addr,
//     conflict-free: consecutive lanes -> consecutive LDS addresses),
//   - nontemporal out-stores / v-loads so the 384 MB of streaming data does
//     not evict the reusable src halo lines from the 192 MB L2,
//   - exact global-gather fallback for samples escaping the halo (|v|>=8,
//     ~1e-15 probability per sample for randn displacements, but required
//     for exact reference semantics).

#define B_   8
#define C_   4
#define D_   32
#define H_   256
#define W_   256
#define TILE 32
#define HALO 8
#define TDIM (TILE + 2 * HALO)        // 48
#define LDS_CH (TDIM * TDIM)          // 2304 floats per channel
#define HW_ (H_ * W_)

__global__ __launch_bounds__(256) void SpatialTransformer_warp2d_kernel(
    const float* __restrict__ src, const float* __restrict__ v,
    float* __restrict__ out) {
  __shared__ __align__(16) float tile[C_ * LDS_CH];  // 36 KB -> 8 blocks/WGP

  const int tid   = threadIdx.x;
  const int tileX = blockIdx.x;
  const int tileY = blockIdx.y;
  const int slice = blockIdx.z;           // b*D + d
  const int b = slice / D_;
  const int d = slice - b * D_;
  const int tx0 = tileX * TILE - HALO;    // tile origin incl. halo
  const int ty0 = tileY * TILE - HALO;
  const int planeBase = b * (C_ * D_) + d;  // plane index for c=0; +c*D_ per ch

  // ---- Stage src tile into LDS: 2304 float4 elements, 256 thr x 9 iters ----
  // x-granularity is 4 floats and tx0 === 0 (mod 4), so every float4 is
  // entirely inside or entirely outside the image -> all-or-none per element.
  const uint32_t ldsBase = (uint32_t)(size_t)(&tile[0]);  // low32 == LDS offset
#pragma unroll
  for (int k = 0; k < 9; ++k) {
    const int i    = tid + k * 256;       // 0..2303
    const int c    = i / 576;
    const int j    = i - c * 576;
    const int row  = j / 12;
    const int col  = (j - row * 12) * 4;  // float index within row, mult of 4
    const int gy   = ty0 + row;
    const int gx   = tx0 + col;
    const int ldsElem = c * LDS_CH + row * TDIM + col;
    if ((unsigned)gy < (unsigned)H_ && (unsigned)gx < (unsigned)W_) {
      const uint32_t ldsAddr = ldsBase + (uint32_t)(ldsElem * 4);
      const uint32_t gOff =
          (uint32_t)((((planeBase + c * D_) * HW_) + gy * W_ + gx) * 4);
      asm volatile("global_load_async_to_lds_b128 %0, %1, %2"
                   :
                   : "v"(ldsAddr), "v"(gOff), "s"(src)
                   : "memory");
    } else {
      *(float4*)(&tile[ldsElem]) = make_float4(0.f, 0.f, 0.f, 0.f);
    }
  }
  asm volatile("s_wait_asynccnt 0" ::: "memory");
  __syncthreads();

  // ---- Sample ----
  const int tx = tid & 31;                 // consecutive lanes -> consecutive x
  const int ty = tid >> 5;                 // 0..7
  const int x  = tileX * TILE + tx;
  const int vPlane = (b * 2 * D_ + d) * HW_;      // v channel 0 plane
  constexpr float SX = (float)(W_ - 1) / (float)(H_ - 1);  // == 1.0 here

#pragma unroll
  for (int r = 0; r < 4; ++r) {
    const int y   = tileY * TILE + ty + r * 8;
    const int pix = y * W_ + x;
    const float vx = __builtin_nontemporal_load(&v[vPlane + pix]);
    const float vy = __builtin_nontemporal_load(&v[vPlane + D_ * HW_ + pix]);

    const float ix  = ((float)x + vx) * SX;
    const float iy  = (float)y + vy;
    const float x0f = floorf(ix);
    const float y0f = floorf(iy);
    const float wx1 = ix - x0f, wx0 = 1.f - wx1;
    const float wy1 = iy - y0f, wy0 = 1.f - wy1;
    const int x0 = (int)x0f, y0 = (int)y0f;
    const int x1 = x0 + 1,  y1 = y0 + 1;
    const float w00 = wx0 * wy0, w10 = wx1 * wy0;
    const float w01 = wx0 * wy1, w11 = wx1 * wy1;

    const int lx = x0 - tx0;
    const int ly = y0 - ty0;
    float acc[C_];

    if ((unsigned)lx < (unsigned)(TDIM - 1) &&
        (unsigned)ly < (unsigned)(TDIM - 1)) {
      // Hot path: all 4 corners inside staged tile (zero padding pre-baked).
#pragma unroll
      for (int c = 0; c < C_; ++c) {
        const float* t = &tile[c * LDS_CH + ly * TDIM + lx];
        float a = t[0] * w00;
        a = fmaf(t[1], w10, a);
        a = fmaf(t[TDIM], w01, a);
        a = fmaf(t[TDIM + 1], w11, a);
        acc[c] = a;
      }
    } else {
      // Cold path (|v| >= HALO): exact reference semantics via global gather.
      const bool vx0 = (x0 >= 0) && (x0 < W_);
      const bool vx1 = (x1 >= 0) && (x1 < W_);
      const bool vy0 = (y0 >= 0) && (y0 < H_);
      const bool vy1 = (y1 >= 0) && (y1 < H_);
      const int cx0 = min(max(x0, 0), W_ - 1);
      const int cx1 = min(max(x1, 0), W_ - 1);
      const int cy0 = min(max(y0, 0), H_ - 1);
      const int cy1 = min(max(y1, 0), H_ - 1);
#pragma unroll
      for (int c = 0; c < C_; ++c) {
        const int base = (planeBase + c * D_) * HW_;
        const float s00 = (vx0 && vy0) ? src[base + cy0 * W_ + cx0] : 0.f;
        const float s10 = (vx1 && vy0) ? src[base + cy0 * W_ + cx1] : 0.f;
        const float s01 = (vx0 && vy1) ? src[base + cy1 * W_ + cx0] : 0.f;
        const float s11 = (vx1 && vy1) ? src[base + cy1 * W_ + cx1] : 0.f;
        float a = s00 * w00;
        a = fmaf(s10, w10, a);
        a = fmaf(s01, w01, a);
        a = fmaf(s11, w11, a);
        acc[c] = a;
      }
    }

    // Single store site; NT so 256 MB of write-once data bypasses L2 reuse.
#pragma unroll
    for (int c = 0; c < C_; ++c) {
      __builtin_nontemporal_store(acc[c], &out[(planeBase + c * D_) * HW_ + pix]);
    }
  }
}

extern "C" void kernel_launch(void* const* d_in, const int* in_sizes, int n_in,
                              void* d_out, int out_size, void* d_ws,
                              size_t ws_size, hipStream_t stream) {
  (void)in_sizes; (void)n_in; (void)out_size; (void)d_ws; (void)ws_size;
  const float* src = (const float*)d_in[0];
  const float* v   = (const float*)d_in[1];
  float* out       = (float*)d_out;
  dim3 grid(W_ / TILE, H_ / TILE, B_ * D_);  // 8 x 8 x 256 blocks
  SpatialTransformer_warp2d_kernel<<<grid, dim3(256), 0, stream>>>(src, v, out);
}